// Model_87136296501727
// MI455X (gfx1250) — compile-verified
//
#include <hip/hip_runtime.h>
#include <hip/hip_bf16.h>
#include <math.h>

typedef float v2f __attribute__((ext_vector_type(2)));
typedef float v8f __attribute__((ext_vector_type(8)));

// ---------------------------------------------------------------------------
// Generic fill
// ---------------------------------------------------------------------------
__global__ void fill_kernel(float* __restrict__ p, float v, long n) {
    long i = (long)blockIdx.x * blockDim.x + threadIdx.x;
    long stride = (long)gridDim.x * blockDim.x;
    for (; i < n; i += stride) p[i] = v;
}

// ---------------------------------------------------------------------------
// C[Nrows x NCOLS] = act( (A [∘ A2]) @ B ),  A:[Nrows,64] row-major,
// B:[64,NCOLS] row-major, K = 64 fixed, NCOLS in {32,64} (compile-time).
// One wave -> one 16x16 tile via v_wmma_f32_16x16x4_f32 (16 steps).
// 256 threads = 8 waves; tiles arranged (8/tilesN) x tilesN.
// B staged transposed in LDS so both frags use ds_load_2addr_b64 with
// immediate offsets; LDS row stride 68 avoids bank conflicts.
// ---------------------------------------------------------------------------
template <int NCOLS>
__global__ __launch_bounds__(256)
void gemm_k64_wmma(const float* __restrict__ A,
                   const float* __restrict__ A2,   // optional elementwise mult
                   const float* __restrict__ B,
                   float* __restrict__ C,
                   int Nrows, int leaky)
{
    constexpr int tilesN = NCOLS / 16;             // 4 (NCOLS=64) or 2 (32)
    constexpr int rowsPB = (8 / tilesN) * 16;      // 32 or 64
    constexpr int LDA    = 68;                     // padded stride (floats)
    constexpr int NSHIFT = (NCOLS == 64) ? 6 : 5;

    __shared__ float As[rowsPB * LDA];
    __shared__ float Bt[NCOLS * LDA];

    const int rowBase = blockIdx.x * rowsPB;

    // Stage B transposed: Bt[n][k] = B[k][n]  (coalesced global read)
    for (int i = threadIdx.x; i < 64 * NCOLS; i += 256) {
        int k = i >> NSHIFT;
        int n = i & (NCOLS - 1);
        Bt[n * LDA + k] = B[i];
    }
    // Stage A rows (fused elementwise multiply), zero-pad OOB rows
    for (int i = threadIdx.x; i < rowsPB * 64; i += 256) {
        int r = rowBase + (i >> 6);
        float v = 0.0f;
        if (r < Nrows) {
            long idx = (long)r * 64 + (i & 63);
            v = A[idx];
            if (A2) v *= A2[idx];
        }
        As[(i >> 6) * LDA + (i & 63)] = v;
    }
    __syncthreads();

    const int wave = threadIdx.x >> 5;
    const int lane = threadIdx.x & 31;
    const int half = lane >> 4;                    // 0: lanes 0-15, 1: 16-31
    const int lm   = lane & 15;
    const int tn   = wave % tilesN;
    const int tm   = wave / tilesN;
    const int rb   = tm << 4;                      // tile row offset in block
    const int cb   = tn << 4;                      // tile col offset

    const float* ap = &As[(rb + lm) * LDA + 2 * half];
    const float* bp = &Bt[(cb + lm) * LDA + 2 * half];

    v8f acc = {};
#pragma unroll
    for (int k = 0; k < 64; k += 4) {
        // A 16x4 f32 frag: lanes 0-15 hold K=k,k+1; lanes 16-31 hold K=k+2,k+3
        v2f a; a.x = ap[k]; a.y = ap[k + 1];
        // B 4x16 f32 frag (from transposed stage): same striping
        v2f b; b.x = bp[k]; b.y = bp[k + 1];
        acc = __builtin_amdgcn_wmma_f32_16x16x4_f32(
            false, a, false, b, (short)0, acc, false, false);
    }

    // C/D layout: VGPR i -> (M = i + 8*half, N = lm)
#pragma unroll
    for (int i = 0; i < 8; ++i) {
        int r = rowBase + rb + i + 8 * half;
        if (r < Nrows) {
            float v = acc[i];
            if (leaky) v = v > 0.0f ? v : 0.01f * v;
            C[(long)r * NCOLS + cb + lm] = v;
        }
    }
}

// ---------------------------------------------------------------------------
// Attention pass for one relation r:
// att[e] = sum_k proj[src[e]][k] * tanh(proj[dst[e]][k] + rel_r[k])
// One wave per edge (grid-stride); lane handles dims k and k+32.
// ---------------------------------------------------------------------------
__global__ __launch_bounds__(256)
void att_pass_kernel(const float* __restrict__ proj,
                     const float* __restrict__ relv,   // 64 floats
                     const int* __restrict__ src,
                     const int* __restrict__ dst,
                     const int* __restrict__ etype,
                     int r, int E,
                     float* __restrict__ att)
{
    int wave = (int)((blockIdx.x * blockDim.x + threadIdx.x) >> 5);
    int nw   = (int)((gridDim.x * blockDim.x) >> 5);
    int lane = threadIdx.x & 31;
    float r0 = relv[lane];
    float r1 = relv[lane + 32];
    for (int e = wave; e < E; e += nw) {
        if (etype[e] != r) continue;
        const float* ps = proj + (long)src[e] * 64;
        const float* pd = proj + (long)dst[e] * 64;
        float p = ps[lane]      * tanhf(pd[lane]      + r0)
                + ps[lane + 32] * tanhf(pd[lane + 32] + r1);
#pragma unroll
        for (int off = 16; off > 0; off >>= 1)
            p += __shfl_xor(p, off, 32);
        if (lane == 0) att[e] = p;
    }
}

// ---------------------------------------------------------------------------
// Segment max via int-trick float atomic max (maps to global_atomic_max/min)
// ---------------------------------------------------------------------------
__global__ void edge_max_kernel(const float* __restrict__ att,
                                const int* __restrict__ dst,
                                float* __restrict__ mx, int E)
{
    int e = blockIdx.x * blockDim.x + threadIdx.x;
    if (e >= E) return;
    float v = att[e];
    float* addr = mx + dst[e];
    if (v >= 0.0f) atomicMax((int*)addr, __float_as_int(v));
    else           atomicMin((unsigned int*)addr, __float_as_uint(v));
}

// w[e] = exp(att[e]-mx[dst]);  denom[dst] += w[e]   (in-place att -> e_w)
__global__ void edge_exp_kernel(float* __restrict__ w,
                                const int* __restrict__ dst,
                                const float* __restrict__ mx,
                                float* __restrict__ denom, int E)
{
    int e = blockIdx.x * blockDim.x + threadIdx.x;
    if (e >= E) return;
    int d = dst[e];
    float ev = expf(w[e] - mx[d]);
    w[e] = ev;
    atomicAdd(denom + d, ev);
}

__global__ void edge_norm_kernel(float* __restrict__ w,
                                 const int* __restrict__ dst,
                                 const float* __restrict__ denom, int E)
{
    int e = blockIdx.x * blockDim.x + threadIdx.x;
    if (e >= E) return;
    w[e] = w[e] / denom[dst[e]];
}

// ---------------------------------------------------------------------------
// agg[dst] += h[src] * w   (thread = one (edge, dim) pair, grid-stride)
// ---------------------------------------------------------------------------
__global__ void aggregate_kernel(const float* __restrict__ h,
                                 const float* __restrict__ w,
                                 const int* __restrict__ src,
                                 const int* __restrict__ dst,
                                 float* __restrict__ agg, long total)
{
    long i = (long)blockIdx.x * blockDim.x + threadIdx.x;
    long stride = (long)gridDim.x * blockDim.x;
    for (; i < total; i += stride) {
        int e = (int)(i >> 6);
        int d = (int)(i & 63);
        float v = h[(long)src[e] * 64 + d] * w[e];
        atomicAdd(agg + (long)dst[e] * 64 + d, v);
    }
}

// ---------------------------------------------------------------------------
// out[n] = [ ent[n] (64) | l2norm(h1[n]) (64) | l2norm(h2[n]) (32) ]
// One wave per node.
// ---------------------------------------------------------------------------
__global__ __launch_bounds__(256)
void finalize_kernel(const float* __restrict__ ent,
                     const float* __restrict__ h1,
                     const float* __restrict__ h2,
                     float* __restrict__ out, int N)
{
    int wave = (int)((blockIdx.x * blockDim.x + threadIdx.x) >> 5);
    int nw   = (int)((gridDim.x * blockDim.x) >> 5);
    int lane = threadIdx.x & 31;
    for (int n = wave; n < N; n += nw) {
        float* o = out + (long)n * 160;
        o[lane]      = ent[(long)n * 64 + lane];
        o[lane + 32] = ent[(long)n * 64 + lane + 32];

        float a0 = h1[(long)n * 64 + lane];
        float a1 = h1[(long)n * 64 + lane + 32];
        float ss = a0 * a0 + a1 * a1;
#pragma unroll
        for (int off = 16; off > 0; off >>= 1) ss += __shfl_xor(ss, off, 32);
        float inv1 = 1.0f / fmaxf(sqrtf(ss), 1e-12f);
        o[64 + lane]      = a0 * inv1;
        o[96 + lane]      = a1 * inv1;

        float b0 = h2[(long)n * 32 + lane];
        float s2 = b0 * b0;
#pragma unroll
        for (int off = 16; off > 0; off >>= 1) s2 += __shfl_xor(s2, off, 32);
        float inv2 = 1.0f / fmaxf(sqrtf(s2), 1e-12f);
        o[128 + lane] = b0 * inv2;
    }
}

// ---------------------------------------------------------------------------
extern "C" void kernel_launch(void* const* d_in, const int* in_sizes, int n_in,
                              void* d_out, int out_size, void* d_ws, size_t ws_size,
                              hipStream_t stream)
{
    const float* ent  = (const float*)d_in[0];   // [N,64]
    const float* rel  = (const float*)d_in[1];   // [R,64]
    const float* W_R  = (const float*)d_in[2];   // [R,64,64]
    const float* W0   = (const float*)d_in[3];   // [64,64]
    const float* W1   = (const float*)d_in[4];   // [64,32]
    const int*   src  = (const int*)d_in[5];     // [E]
    const int*   dst  = (const int*)d_in[6];     // [E]
    const int*   et   = (const int*)d_in[7];     // [E]
    float*       out  = (float*)d_out;

    const int N = in_sizes[0] / 64;
    const int R = in_sizes[2] / (64 * 64);
    const int E = in_sizes[5];

    // Workspace layout (floats): proj | w | mx | denom | agg | h1 | h2
    float* ws   = (float*)d_ws;
    float* proj = ws;                 ws += (long)N * 64;
    float* w    = ws;                 ws += E;
    float* mx   = ws;                 ws += N;
    float* den  = ws;                 ws += N;
    float* agg  = ws;                 ws += (long)N * 64;
    float* h1   = ws;                 ws += (long)N * 64;
    float* h2   = ws;                 ws += (long)N * 32;
    (void)ws_size; (void)n_in; (void)out_size;

    const int T = 256;
    // init softmax state
    fill_kernel<<<1024, T, 0, stream>>>(mx, -__builtin_huge_valf(), (long)N);
    fill_kernel<<<1024, T, 0, stream>>>(den, 0.0f, (long)N);

    // ---- attention: one relation at a time (proj buffer is [N,64]) ----
    const int gemmBlocks64 = (N + 31) / 32;
    for (int r = 0; r < R; ++r) {
        gemm_k64_wmma<64><<<gemmBlocks64, T, 0, stream>>>(
            ent, nullptr, W_R + (long)r * 64 * 64, proj, N, 0);
        att_pass_kernel<<<2048, T, 0, stream>>>(
            proj, rel + (long)r * 64, src, dst, et, r, E, w);
    }

    // ---- edge softmax ----
    const int eb = (E + T - 1) / T;
    edge_max_kernel<<<eb, T, 0, stream>>>(w, dst, mx, E);
    edge_exp_kernel<<<eb, T, 0, stream>>>(w, dst, mx, den, E);
    edge_norm_kernel<<<eb, T, 0, stream>>>(w, dst, den, E);

    // ---- layer 1: agg = segsum(ent[src]*w); h1 = leaky((ent*agg)@W0) ----
    fill_kernel<<<2048, T, 0, stream>>>(agg, 0.0f, (long)N * 64);
    aggregate_kernel<<<4096, T, 0, stream>>>(ent, w, src, dst, agg, (long)E * 64);
    gemm_k64_wmma<64><<<gemmBlocks64, T, 0, stream>>>(ent, agg, W0, h1, N, 1);

    // ---- layer 2: agg = segsum(h1[src]*w); h2 = leaky((h1*agg)@W1) ----
    fill_kernel<<<2048, T, 0, stream>>>(agg, 0.0f, (long)N * 64);
    aggregate_kernel<<<4096, T, 0, stream>>>(h1, w, src, dst, agg, (long)E * 64);
    const int gemmBlocks32 = (N + 63) / 64;
    gemm_k64_wmma<32><<<gemmBlocks32, T, 0, stream>>>(h1, agg, W1, h2, N, 1);

    // ---- concat + l2norm ----
    finalize_kernel<<<(N + 7) / 8, T, 0, stream>>>(ent, h1, h2, out, N);
}